// NettackGCN_59596966199899
// MI455X (gfx1250) — compile-verified
//
#include <hip/hip_runtime.h>

typedef __attribute__((ext_vector_type(2))) float v2f;
typedef __attribute__((ext_vector_type(8))) float v8f;

// ---------------- norm precompute ----------------

__global__ void set_deg_kernel(float* __restrict__ deg, int n) {
    int i = blockIdx.x * blockDim.x + threadIdx.x;
    if (i < n) deg[i] = 1.0f;   // self-loop weight (fill_value = 1.0)
}

__global__ void accum_deg_kernel(const long long* __restrict__ ei,
                                 const float* __restrict__ w,
                                 float* __restrict__ deg, long long E) {
    long long e = blockIdx.x * (long long)blockDim.x + threadIdx.x;
    if (e < E) atomicAdd(&deg[ei[E + e]], w[e]);   // segment_sum by col
}

__global__ void dinv_kernel(float* __restrict__ deg, int n) {
    int i = blockIdx.x * blockDim.x + threadIdx.x;
    if (i < n) {
        float d = deg[i];
        deg[i] = (d > 0.0f) ? rsqrtf(d) : 0.0f;
    }
}

// ---------------- fp32 WMMA GEMM: C[MxN] = A[MxK_] @ B[K_xN] ----------------
// Block = 256 threads = 8 wave32s; each wave owns one 16-row tile and ALL NT_
// column tiles (one A load feeds NT_ WMMAs). B is staged in LDS *transposed*
// ([col][K], K-stride padded to KC+4 dwords) so each B operand is a single
// aligned ds_load_b64 at an immediate offset, with conflict-free banking
// (stride % 64 == 4 spreads the 32 b64 lanes over all 64 banks).
// A is software-prefetched one k-step ahead to hide global latency behind the
// NT_ WMMAs of the current step.
//
// Requirements: M % 16 == 0, K_ % 4 == 0, A row stride K_, B row stride N,
// C row stride N. The 'active' guard is wave-uniform and forced scalar via
// readfirstlane (WMMA requires EXEC all-ones; never predicate it).
//
// V_WMMA_F32_16X16X4_F32 operand layout:
//   A 16x4 : lanes 0-15 (row=lane) hold {k,k+1}; lanes 16-31 hold {k+2,k+3}
//   B 4x16 : lanes 0-15 (col=lane) hold rows {k,k+1}; lanes 16-31 {k+2,k+3}
//   D 16x16: VGPR v -> row v (lanes 0-15) / row v+8 (lanes 16-31)
template <int K_, int NT_>
__global__ __launch_bounds__(256) void wmma_gemm_kernel(
    const float* __restrict__ A, const float* __restrict__ B,
    float* __restrict__ C, int M, int N)
{
    constexpr int NC   = NT_ * 16;
    constexpr int KC   = (K_ * NC * 4 > 32768) ? (K_ / 2) : K_;  // K rows per LDS chunk
    constexpr int LDSK = KC + 4;                                 // padded K stride (dwords)
    __shared__ __align__(16) float lds_b[NC * LDSK];

    const int wavei = __builtin_amdgcn_readfirstlane(threadIdx.x >> 5);
    const int lane  = threadIdx.x & 31;
    const int tm    = blockIdx.x * 8 + wavei;
    const bool active = (tm * 16) < M;

    const int half = lane >> 4;     // 0: K pair {k,k+1}, 1: {k+2,k+3}
    const int l16  = lane & 15;
    const int arow = tm * 16 + l16;
    const float* ap = A + (size_t)arow * K_ + 2 * half;

    // per-column-tile LDS base: column (t*16+l16), K offset 2*half
    const float* bp[NT_];
#pragma unroll
    for (int t = 0; t < NT_; ++t)
        bp[t] = &lds_b[(t * 16 + l16) * LDSK + 2 * half];

    v8f acc[NT_];
#pragma unroll
    for (int t = 0; t < NT_; ++t) acc[t] = (v8f){};

    for (int k0 = 0; k0 < K_; k0 += KC) {
        __syncthreads();   // protect LDS reads of the previous chunk
        // stage chunk [k0, k0+KC): coalesced global read, transposed LDS write.
        // KC*NC is a multiple of 256 -> no tail.
#pragma unroll 4
        for (int idx = threadIdx.x; idx < KC * NC; idx += 256) {
            int kk = idx / NC, c = idx - kk * NC;
            lds_b[c * LDSK + kk] = (c < N) ? B[(k0 + kk) * N + c] : 0.0f;
        }
        __syncthreads();

        if (active) {
            v2f a = *(const v2f*)(ap + k0);          // first A fragment of chunk
#pragma unroll
            for (int k = 0; k < KC; k += 4) {
                v2f an = {};
                if (k + 4 < KC)                      // compile-time after unroll
                    an = *(const v2f*)(ap + k0 + k + 4);   // prefetch next step
#pragma unroll
                for (int t = 0; t < NT_; ++t) {
                    v2f b = *(const v2f*)(bp[t] + k);      // ds_load_b64, imm offset
                    acc[t] = __builtin_amdgcn_wmma_f32_16x16x4_f32(
                        /*neg_a=*/false, a, /*neg_b=*/false, b,
                        /*c_mod=*/(short)0, acc[t],
                        /*reuse_a=*/false, /*reuse_b=*/false);
                }
                a = an;
            }
        }
    }

    if (active) {
#pragma unroll
        for (int t = 0; t < NT_; ++t) {
            int col = t * 16 + l16;
            if (col < N) {
#pragma unroll
                for (int v = 0; v < 8; ++v) {
                    int row = tm * 16 + half * 8 + v;
                    C[(size_t)row * N + col] = acc[t][v];
                }
            }
        }
    }
}

// ---------------- self-loop + bias init: out[i,f] = bias[f] + dinv[i]^2 * h[i,f] ----------------

__global__ void self_bias_kernel(const float* __restrict__ h,
                                 const float* __restrict__ dinv,
                                 const float* __restrict__ bias,
                                 float* __restrict__ out,
                                 unsigned n, unsigned F) {
    unsigned idx = blockIdx.x * blockDim.x + threadIdx.x;
    if (idx >= n * F) return;
    unsigned i = idx / F;
    unsigned f = idx - i * F;
    float di = dinv[i];
    out[idx] = bias[f] + di * di * h[idx];
}

// ---------------- edge message passing: out[col] += norm * h[row] ----------------
// TPE threads per edge, each handles a float4 feature chunk. ld = feature stride (mult of 4).

template <int TPE>
__global__ void edge_mp_kernel(const long long* __restrict__ ei,
                               const float* __restrict__ w,
                               const float* __restrict__ dinv,
                               const float* __restrict__ h,
                               float* __restrict__ out,
                               long long E, int ld) {
    long long gid = blockIdx.x * (long long)blockDim.x + threadIdx.x;
    long long e = gid / TPE;
    int t = (int)(gid - e * TPE);
    if (e >= E) return;
    long long r = ei[e];
    long long c = ei[E + e];
    float coeff = dinv[r] * w[e] * dinv[c];
    const float4 hv = *(const float4*)(h + r * (long long)ld + t * 4);
    float* op = out + c * (long long)ld + t * 4;
    atomicAdd(op + 0, coeff * hv.x);
    atomicAdd(op + 1, coeff * hv.y);
    atomicAdd(op + 2, coeff * hv.z);
    atomicAdd(op + 3, coeff * hv.w);
}

// ---------------- launcher ----------------

extern "C" void kernel_launch(void* const* d_in, const int* in_sizes, int n_in,
                              void* d_out, int out_size, void* d_ws, size_t ws_size,
                              hipStream_t stream) {
    const float*     x  = (const float*)d_in[0];
    const long long* ei = (const long long*)d_in[1];   // int64 [2, E]
    const float*     ew = (const float*)d_in[2];
    const float*     W1 = (const float*)d_in[3];
    const float*     b1 = (const float*)d_in[4];
    const float*     W2 = (const float*)d_in[5];
    const float*     b2 = (const float*)d_in[6];
    float* out = (float*)d_out;

    const int       H = in_sizes[4];            // 64
    const int       C = in_sizes[6];            // 40
    const int       F = in_sizes[3] / H;        // 256
    const int       N = in_sizes[0] / F;        // 100000
    const long long E = (long long)in_sizes[1] / 2;  // 3,200,000

    // workspace carve-out (256B aligned slabs)
    char* ws = (char*)d_ws;
    size_t off = 0;
    auto carve = [&](size_t bytes) -> float* {
        float* p = (float*)(ws + off);
        off += (bytes + 255) & ~(size_t)255;
        return p;
    };
    float* dinv = carve((size_t)N * sizeof(float));
    float* h1   = carve((size_t)N * H * sizeof(float));
    float* o1   = carve((size_t)N * H * sizeof(float));
    float* h2   = carve((size_t)N * C * sizeof(float));

    const int B = 256;  // 8 wave32s per block

    // 1) degree + dinv (in place)
    set_deg_kernel<<<(N + B - 1) / B, B, 0, stream>>>(dinv, N);
    accum_deg_kernel<<<(int)((E + B - 1) / B), B, 0, stream>>>(ei, ew, dinv, E);
    dinv_kernel<<<(N + B - 1) / B, B, 0, stream>>>(dinv, N);

    const int gemm_blocks = (N + 127) / 128;    // 8 row-tiles per block

    // 2) layer 1 projection: h1 = x @ W1   [N x 256] @ [256 x 64]
    wmma_gemm_kernel<256, 4><<<gemm_blocks, B, 0, stream>>>(x, W1, h1, N, H);

    // 3) layer 1 aggregate: o1 = b1 + dinv^2*h1 (self) + scatter(norm * h1[row])
    self_bias_kernel<<<(int)(((long long)N * H + B - 1) / B), B, 0, stream>>>(h1, dinv, b1, o1, N, H);
    edge_mp_kernel<16><<<(int)((E * 16 + B - 1) / B), B, 0, stream>>>(ei, ew, dinv, h1, o1, E, H);

    // 4) layer 2 projection: h2 = o1 @ W2   [N x 64] @ [64 x 40]
    wmma_gemm_kernel<64, 3><<<gemm_blocks, B, 0, stream>>>(o1, W2, h2, N, C);

    // 5) layer 2 aggregate into d_out
    self_bias_kernel<<<(int)(((long long)N * C + B - 1) / B), B, 0, stream>>>(h2, dinv, b2, out, N, C);
    edge_mp_kernel<10><<<(int)((E * 10 + B - 1) / B), B, 0, stream>>>(ei, ew, dinv, h2, out, E, C);
}